// GraphConv_4707284157012
// MI455X (gfx1250) — compile-verified
//
#include <hip/hip_runtime.h>
#include <hip/hip_bf16.h>

// GraphConv-style sparse scatter:  out[r, b] += weight[etype[e]] * inp[cols[e], b]
//   inp[c, b] = x[b, c]  (x viewed as [B, S*FIN]),  out is [S*FOUT, B],
//   d_out[b*NR + r] = out[r, b].
//
// Memory/atomic bound (256 MFLOP vs ~300MB traffic; both the 51.2MB source
// table and 51.2MB accumulator are L2-resident on MI455X's 192MB L2).
// No dense contraction -> WMMA cannot help; optimize bytes & issue rate.

typedef float v4f __attribute__((ext_vector_type(4)));

#define TPB 256

// ---------------------------------------------------------------- zero scratch
__global__ __launch_bounds__(TPB) void zero_f4_kernel(v4f* __restrict__ p, long n4) {
    long i = (long)blockIdx.x * blockDim.x + threadIdx.x;
    if (i < n4) {
        v4f z = {0.f, 0.f, 0.f, 0.f};
        p[i] = z;
    }
}

// ------------------------------------------------- x [B, NC] -> tx [NC, B=16]
// Reads: for fixed b, consecutive lanes hit consecutive addresses (coalesced).
// Writes: 4x b128 per thread, 64B-contiguous per thread. x is read exactly once
// here -> non-temporal loads keep it from polluting near caches.
__global__ __launch_bounds__(TPB) void transpose_in_kernel(const float* __restrict__ x,
                                                           float* __restrict__ tx,
                                                           long NC) {
    long c = (long)blockIdx.x * blockDim.x + threadIdx.x;
    if (c >= NC) return;
    float v[16];
#pragma unroll
    for (int b = 0; b < 16; ++b)
        v[b] = __builtin_nontemporal_load(x + (long)b * NC + c);
    v4f* dst = (v4f*)(tx + c * 16);
#pragma unroll
    for (int q = 0; q < 4; ++q) {
        v4f t = {v[4 * q + 0], v[4 * q + 1], v[4 * q + 2], v[4 * q + 3]};
        dst[q] = t;  // regular TH: re-read many times from L2 by edge kernel
    }
}

// ------------------------------------------------------------------ edge loop
// 4 lanes per edge, each lane owns one 16B quarter of the edge's 64B row:
//   - one global_load_b128 gathers 8 edges' worth per wave (full line use)
//   - 4 independent no-return global_atomic_add_f32 per lane into a
//     contiguous 16B quarter of the L2-resident accumulator line
//   - index streams are read-once -> NT loads + global_prefetch_b8 run-ahead
__global__ __launch_bounds__(TPB) void edge_kernel(const float* __restrict__ tx,
                                                   const float* __restrict__ weight,
                                                   const int* __restrict__ rows,
                                                   const int* __restrict__ cols,
                                                   const int* __restrict__ etype,
                                                   float* __restrict__ acc,
                                                   int E) {
    long tid = (long)blockIdx.x * blockDim.x + threadIdx.x;
    int e = (int)(tid >> 2);
    int q = (int)(tid & 3);  // 16B quarter of the 64B row
    if (e >= E) return;

    // Speculative prefetch of the index streams ~32KB ahead (one lane / 8 waves).
    if ((threadIdx.x & 255) == 0) {
        __builtin_prefetch(rows + e + 8192, 0, 1);
        __builtin_prefetch(cols + e + 8192, 0, 1);
        __builtin_prefetch(etype + e + 8192, 0, 1);
    }

    int r = __builtin_nontemporal_load(rows + e);
    int c = __builtin_nontemporal_load(cols + e);
    int t = __builtin_nontemporal_load(etype + e);
    float w = weight[t];  // 256B table, L1-resident

    const v4f g = *(const v4f*)(tx + (long)c * 16 + q * 4);  // b128 gather (L2 hit)
    float* dst = acc + (long)r * 16 + q * 4;
    unsafeAtomicAdd(dst + 0, w * g.x);  // native f32 atomics, no return
    unsafeAtomicAdd(dst + 1, w * g.y);
    unsafeAtomicAdd(dst + 2, w * g.z);
    unsafeAtomicAdd(dst + 3, w * g.w);
}

// --------------------------------------------- acc [NR, 16] -> d_out [16, NR]
// Reads: 4x b128 per thread (contiguous). Writes: per b, consecutive lanes hit
// consecutive addresses (coalesced); output never re-read -> NT stores.
__global__ __launch_bounds__(TPB) void transpose_out_kernel(const float* __restrict__ acc,
                                                            float* __restrict__ out,
                                                            long NR) {
    long r = (long)blockIdx.x * blockDim.x + threadIdx.x;
    if (r >= NR) return;
    const v4f* src = (const v4f*)(acc + r * 16);
    float v[16];
#pragma unroll
    for (int q = 0; q < 4; ++q) {
        v4f t = src[q];
        v[4 * q + 0] = t.x;
        v[4 * q + 1] = t.y;
        v[4 * q + 2] = t.z;
        v[4 * q + 3] = t.w;
    }
#pragma unroll
    for (int b = 0; b < 16; ++b)
        __builtin_nontemporal_store(v[b], out + (long)b * NR + r);
}

// ------------------------------------------------ fallback (no workspace path)
__global__ __launch_bounds__(TPB) void edge_direct_kernel(const float* __restrict__ x,
                                                          const float* __restrict__ weight,
                                                          const int* __restrict__ rows,
                                                          const int* __restrict__ cols,
                                                          const int* __restrict__ etype,
                                                          float* __restrict__ out,
                                                          int E, long NC, long NR) {
    long tid = (long)blockIdx.x * blockDim.x + threadIdx.x;
    int e = (int)(tid >> 4);
    int b = (int)(tid & 15);
    if (e >= E) return;
    int r = __builtin_nontemporal_load(rows + e);
    int c = __builtin_nontemporal_load(cols + e);
    int t = __builtin_nontemporal_load(etype + e);
    float v = weight[t] * x[(long)b * NC + c];
    unsafeAtomicAdd(out + (long)b * NR + r, v);
}

extern "C" void kernel_launch(void* const* d_in, const int* in_sizes, int n_in,
                              void* d_out, int out_size, void* d_ws, size_t ws_size,
                              hipStream_t stream) {
    const float* x      = (const float*)d_in[0];  // [B, S, FIN] f32
    const float* weight = (const float*)d_in[1];  // [T] f32
    const int*   rows   = (const int*)d_in[2];    // [E] i32
    const int*   cols   = (const int*)d_in[3];    // [E] i32
    const int*   etype  = (const int*)d_in[4];    // [E] i32
    float* out = (float*)d_out;

    const int B = 16, FIN = 16, FOUT = 16;
    const int E = in_sizes[2];
    const long S  = (long)in_sizes[0] / ((long)B * FIN);
    const long NC = S * FIN;   // source index space  (cols in [0, NC))
    const long NR = S * FOUT;  // dest   index space  (rows in [0, NR))

    const size_t need = (size_t)(NC + NR) * B * sizeof(float);

    if (ws_size >= need) {
        float* tx  = (float*)d_ws;       // [NC, 16]
        float* acc = tx + NC * B;        // [NR, 16]

        long n4 = (NR * B) / 4;
        zero_f4_kernel<<<(int)((n4 + TPB - 1) / TPB), TPB, 0, stream>>>((v4f*)acc, n4);

        transpose_in_kernel<<<(int)((NC + TPB - 1) / TPB), TPB, 0, stream>>>(x, tx, NC);

        const long edge_threads = (long)E * 4;  // 4 lanes per edge
        edge_kernel<<<(int)((edge_threads + TPB - 1) / TPB), TPB, 0, stream>>>(
            tx, weight, rows, cols, etype, acc, E);

        transpose_out_kernel<<<(int)((NR + TPB - 1) / TPB), TPB, 0, stream>>>(acc, out, NR);
    } else {
        // Workspace too small: accumulate directly into d_out (strided layout).
        long n4 = (long)out_size / 4;
        zero_f4_kernel<<<(int)((n4 + TPB - 1) / TPB), TPB, 0, stream>>>((v4f*)out, n4);

        const long edge_threads = (long)E * 16;
        edge_direct_kernel<<<(int)((edge_threads + TPB - 1) / TPB), TPB, 0, stream>>>(
            x, weight, rows, cols, etype, out, E, NC, NR);
    }
}